// GlobalCoC_10754598109696
// MI455X (gfx1250) — compile-verified
//
#include <hip/hip_runtime.h>
#include <math.h>

// ---------------------------------------------------------------------------
// Problem constants (from setup_inputs): n=4, l=4800, s=300, c=256, hd=256, fc=8
// ---------------------------------------------------------------------------
#define N_  4
#define L_  4800
#define S_  300
#define C_  256
#define HD_ 256
#define FC_ 8
#define SC_ 32           // hd / fc

typedef float v2f __attribute__((ext_vector_type(2)));
typedef float v8f __attribute__((ext_vector_type(8)));

__device__ __forceinline__ float gelu_exact(float v) {
    return 0.5f * v * (1.0f + erff(v * 0.7071067811865475f));
}

// ---------------------------------------------------------------------------
// fp32 WMMA fragment layouts (CDNA5 ISA 7.12.2), wave32:
//   A 16x4 : lanes 0-15 hold row M=lane, k={k0,k0+1}; lanes 16-31 hold k={k0+2,k0+3}
//   B 4x16 : lanes 0-15 hold col N=lane, k rows {k0,k0+1}; lanes 16-31 rows {k0+2,k0+3}
//   C 16x16: VGPR v, lanes 0-15 -> (M=v, N=lane); lanes 16-31 -> (M=v+8, N=lane-16)
// ---------------------------------------------------------------------------

// ---------------------------------------------------------------------------
// Register-tiled fp32 WMMA GEMM, all shapes compile-time.
// One wave computes MT*16 x NT*16 (64x32, 8 accumulators = 64 VGPRs, no
// spills). Single walking A / B pointers; all subtile strides are immediate
// offsets. Virtual concat along K: A1 supplies k >= KSPLIT (compile-time
// two-phase loop).
// ---------------------------------------------------------------------------
template <int MT, int NT, int LDA0, int LDA1, int KSPLIT, int LDB, int LDC,
          int K, int GELU>
__global__ void gemm_wmma(const float* __restrict__ A0,
                          const float* __restrict__ A1,
                          const float* __restrict__ B,
                          const float* __restrict__ bias,
                          float* __restrict__ C) {
    const int m0   = blockIdx.x * (MT * 16);
    const int n0   = blockIdx.y * (NT * 16);
    const int lane = threadIdx.x;
    const int l16  = lane & 15;
    const int koff = (lane >> 4) << 1;     // 0 for lanes 0-15, 2 for lanes 16-31

    v8f acc[MT][NT];
    #pragma unroll
    for (int mt = 0; mt < MT; ++mt)
        #pragma unroll
        for (int nt = 0; nt < NT; ++nt) acc[mt][nt] = (v8f){};

    const float* Ap = A0 + (size_t)(m0 + l16) * LDA0 + koff;
    const float* Bp = B + (size_t)koff * LDB + n0 + l16;

    #pragma unroll 2
    for (int kg = 0; kg < KSPLIT; kg += 4) {
        v2f a[MT], b[NT];
        #pragma unroll
        for (int mt = 0; mt < MT; ++mt) {      // contiguous pair -> b64 load
            a[mt].x = Ap[mt * 16 * LDA0];
            a[mt].y = Ap[mt * 16 * LDA0 + 1];
        }
        #pragma unroll
        for (int nt = 0; nt < NT; ++nt) {
            b[nt].x = Bp[nt * 16];
            b[nt].y = Bp[LDB + nt * 16];
        }
        #pragma unroll
        for (int mt = 0; mt < MT; ++mt)
            #pragma unroll
            for (int nt = 0; nt < NT; ++nt)
                acc[mt][nt] = __builtin_amdgcn_wmma_f32_16x16x4_f32(
                    false, a[mt], false, b[nt], (short)0, acc[mt][nt],
                    false, false);
        Ap += 4;
        Bp += 4 * LDB;
    }

    if constexpr (KSPLIT < K) {
        const float* Aq = A1 + (size_t)(m0 + l16) * LDA1 + koff;
        #pragma unroll 2
        for (int kg = KSPLIT; kg < K; kg += 4) {
            v2f a[MT], b[NT];
            #pragma unroll
            for (int mt = 0; mt < MT; ++mt) {
                a[mt].x = Aq[mt * 16 * LDA1];
                a[mt].y = Aq[mt * 16 * LDA1 + 1];
            }
            #pragma unroll
            for (int nt = 0; nt < NT; ++nt) {
                b[nt].x = Bp[nt * 16];
                b[nt].y = Bp[LDB + nt * 16];
            }
            #pragma unroll
            for (int mt = 0; mt < MT; ++mt)
                #pragma unroll
                for (int nt = 0; nt < NT; ++nt)
                    acc[mt][nt] = __builtin_amdgcn_wmma_f32_16x16x4_f32(
                        false, a[mt], false, b[nt], (short)0, acc[mt][nt],
                        false, false);
            Aq += 4;
            Bp += 4 * LDB;
        }
    }

    const int rhalf = (lane >> 4) << 3;    // +0 or +8
    #pragma unroll
    for (int nt = 0; nt < NT; ++nt) {
        const int col = n0 + nt * 16 + l16;
        const float bv = bias[col];
        #pragma unroll
        for (int mt = 0; mt < MT; ++mt) {
            const int rbase = m0 + mt * 16 + rhalf;
            #pragma unroll
            for (int v = 0; v < 8; ++v) {
                float o = acc[mt][nt][v] + bv;
                if (GELU) o = gelu_exact(o);
                C[(size_t)(rbase + v) * LDC + col] = o;
            }
        }
    }
}

// ---------------------------------------------------------------------------
// Small fp32 WMMA GEMM (16x16 tile per wave) for M not a multiple of 64
// (center projection, M=1200).
// ---------------------------------------------------------------------------
template <int LDA, int LDB, int LDC, int K>
__global__ void gemm_wmma_small(const float* __restrict__ A,
                                const float* __restrict__ B,
                                const float* __restrict__ bias,
                                float* __restrict__ C) {
    const int m0   = blockIdx.x * 16;
    const int n0   = blockIdx.y * 16;
    const int lane = threadIdx.x;
    const int l16  = lane & 15;
    const int koff = (lane >> 4) << 1;
    const float* Ap = A + (size_t)(m0 + l16) * LDA + koff;
    const float* Bp = B + (size_t)koff * LDB + n0 + l16;

    v8f acc = {};
    #pragma unroll 2
    for (int k0 = 0; k0 < K; k0 += 4) {
        v2f a;  a.x = Ap[0];  a.y = Ap[1];
        v2f b;  b.x = Bp[0];  b.y = Bp[LDB];
        acc = __builtin_amdgcn_wmma_f32_16x16x4_f32(
                  false, a, false, b, (short)0, acc, false, false);
        Ap += 4;
        Bp += 4 * LDB;
    }
    const float bv    = bias[n0 + l16];
    const int   rbase = m0 + ((lane >> 4) << 3);
    #pragma unroll
    for (int v = 0; v < 8; ++v)
        C[(size_t)(rbase + v) * LDC + (n0 + l16)] = acc[v] + bv;
}

// ---------------------------------------------------------------------------
// 3x3 SAME conv (NHWC, 512 -> 256) as implicit-im2col WMMA GEMM.
// One wave per (16 pixels x 64 out-channels) tile.
// Boundary handling is branch-free: out-of-image taps clamp the load address
// to the lane's own (always valid) pixel and multiply the fragment by a
// {0,1} mask -> unconditional contiguous b64 A-loads, no exec-mask churn.
// ---------------------------------------------------------------------------
__global__ void conv3x3_wmma_f32(const float* __restrict__ Y,    // (n, h, w, 512)
                                 const float* __restrict__ Wc,   // (3, 3, 512, 256)
                                 const float* __restrict__ bc,   // (256)
                                 float* __restrict__ O,          // (n, h*w, 256)
                                 const int* __restrict__ hp,
                                 const int* __restrict__ wp) {
    const int h  = *hp;
    const int w  = *wp;
    const int hw = h * w;

    const int m0   = blockIdx.x * 16;
    const int oc0  = blockIdx.y * 64;
    const int lane = threadIdx.x;
    const int l16  = lane & 15;
    const int koff = (lane >> 4) << 1;

    // decompose this lane's pixel index (always in-bounds)
    const int m  = m0 + l16;
    const int ni = m / hw;
    const int r  = m - ni * hw;
    const int yi = r / w;
    const int xi = r - yi * w;
    const float* yself = Y + ((size_t)ni * hw + (size_t)yi * w + xi) * 512 + koff;

    v8f acc[4];
    #pragma unroll
    for (int nt = 0; nt < 4; ++nt) acc[nt] = (v8f){};

    #pragma unroll
    for (int ky = 0; ky < 3; ++ky) {
        #pragma unroll
        for (int kx = 0; kx < 3; ++kx) {
            const int  yy    = yi + ky - 1;
            const int  xx    = xi + kx - 1;
            const bool valid = (yy >= 0) && (yy < h) && (xx >= 0) && (xx < w);
            const float     msk = valid ? 1.0f : 0.0f;
            const ptrdiff_t dlt = valid ? (ptrdiff_t)((ky - 1) * w + (kx - 1)) * 512
                                        : (ptrdiff_t)0;
            const float* yb = yself + dlt;
            const float* wb = Wc + (size_t)((ky * 3 + kx) * 512) * 256 + oc0 + l16
                                 + (size_t)koff * 256;

            #pragma unroll 2
            for (int k0 = 0; k0 < 512; k0 += 4) {
                v2f a;                              // one b64 load, then mask
                a.x = yb[0] * msk;
                a.y = yb[1] * msk;
                v2f b[4];
                #pragma unroll
                for (int nt = 0; nt < 4; ++nt) {
                    b[nt].x = wb[nt * 16];
                    b[nt].y = wb[256 + nt * 16];
                }
                #pragma unroll
                for (int nt = 0; nt < 4; ++nt)
                    acc[nt] = __builtin_amdgcn_wmma_f32_16x16x4_f32(
                        false, a, false, b[nt], (short)0, acc[nt], false, false);
                yb += 4;
                wb += 4 * 256;
            }
        }
    }

    const int rbase = m0 + ((lane >> 4) << 3);
    #pragma unroll
    for (int nt = 0; nt < 4; ++nt) {
        const int col = oc0 + nt * 16 + l16;
        const float bv = bc[col];
        #pragma unroll
        for (int v = 0; v < 8; ++v)
            O[(size_t)(rbase + v) * 256 + col] = acc[nt][v] + bv;
    }
}

// ---------------------------------------------------------------------------
// Similarity + argmax + dispatch. sigmoid is monotonic => argmax over s of
// sigmoid(a*sim+b) == argmax of raw normalized dot. Normalized c_point cached
// in LDS (300x32 f32 = 38.4 KB); x-row normalization folded into the winning
// value (positive scale doesn't change argmax).
// Writes disp in (n, l, fc*sc) layout (== reference transpose+reshape).
// ---------------------------------------------------------------------------
__global__ void sim_dispatch(const float* __restrict__ xp,   // (n, l, 256)
                             const float* __restrict__ cp,   // (n, s, 512)
                             const float* __restrict__ alpha,
                             const float* __restrict__ beta,
                             float* __restrict__ disp) {     // (n, l, 256)
    __shared__ float sP[S_ * SC_];

    const int bf  = blockIdx.y;
    const int ni  = bf / FC_;
    const int f   = bf % FC_;
    const int tid = threadIdx.x;

    // stage this head's c_point rows into LDS
    for (int idx = tid; idx < S_ * SC_; idx += blockDim.x) {
        const int si = idx / SC_;
        const int j  = idx - si * SC_;
        sP[idx] = cp[((size_t)(ni * S_ + si)) * (2 * HD_) + f * (2 * SC_) + j];
    }
    __syncthreads();

    // l2-normalize rows in place
    for (int si = tid; si < S_; si += blockDim.x) {
        float ss = 0.0f;
        #pragma unroll
        for (int j = 0; j < SC_; ++j) { float v = sP[si * SC_ + j]; ss += v * v; }
        const float inv = 1.0f / fmaxf(sqrtf(ss), 1e-12f);
        #pragma unroll
        for (int j = 0; j < SC_; ++j) sP[si * SC_ + j] *= inv;
    }
    __syncthreads();

    const int li = blockIdx.x * blockDim.x + tid;
    if (li >= L_) return;

    float xv[SC_];
    float ss = 0.0f;
    const size_t xbase = ((size_t)(ni * L_ + li)) * HD_ + f * SC_;
    #pragma unroll
    for (int j = 0; j < SC_; ++j) { xv[j] = xp[xbase + j]; ss += xv[j] * xv[j]; }
    const float xinv = 1.0f / fmaxf(sqrtf(ss), 1e-12f);

    float best = -3.4e38f;
    int   bi   = 0;
    for (int si = 0; si < S_; ++si) {
        float d = 0.0f;
        #pragma unroll
        for (int j = 0; j < SC_; ++j) d += xv[j] * sP[si * SC_ + j];
        if (d > best) { best = d; bi = si; }
    }
    best *= xinv;

    const float t    = alpha[0] * best + beta[0];
    const float sval = 1.0f / (1.0f + __expf(-t));

    // gather winning c_value row from global and scale
    const size_t vbase = ((size_t)(ni * S_ + bi)) * (2 * HD_) + f * (2 * SC_) + SC_;
    const size_t obase = (((size_t)(ni * L_ + li)) * FC_ + f) * SC_;
    #pragma unroll
    for (int j = 0; j < SC_; ++j)
        disp[obase + j] = sval * cp[vbase + j];
}

// ---------------------------------------------------------------------------
// LayerNorm over last dim C_=256 (+ optional residual). One 256-thread block
// per token.
// ---------------------------------------------------------------------------
__global__ void layernorm_k(const float* __restrict__ X,
                            const float* __restrict__ g,
                            const float* __restrict__ b,
                            const float* __restrict__ resid,   // may be null
                            float* __restrict__ Out) {
    __shared__ float r1[C_];
    __shared__ float r2[C_];
    const int row = blockIdx.x;
    const int tid = threadIdx.x;

    const float v = X[(size_t)row * C_ + tid];
    r1[tid] = v;
    r2[tid] = v * v;
    __syncthreads();
    for (int off = C_ / 2; off > 0; off >>= 1) {
        if (tid < off) { r1[tid] += r1[tid + off]; r2[tid] += r2[tid + off]; }
        __syncthreads();
    }
    const float mu  = r1[0] * (1.0f / C_);
    const float var = r2[0] * (1.0f / C_) - mu * mu;
    const float rs  = rsqrtf(var + 1e-5f);

    float o = (v - mu) * rs * g[tid] + b[tid];
    if (resid) o += resid[(size_t)row * C_ + tid];
    Out[(size_t)row * C_ + tid] = o;
}

// ---------------------------------------------------------------------------
// Host side
// ---------------------------------------------------------------------------
static void run_block(const float* x, const float* center,
                      const int* hp, const int* wp,
                      void* const* d_in, float* out_slice, float* ws,
                      hipStream_t stream) {
    const float* Wp0 = (const float*)d_in[8];
    const float* bp0 = (const float*)d_in[9];
    const float* Wp1 = (const float*)d_in[10];
    const float* bp1 = (const float*)d_in[11];
    const float* Wm  = (const float*)d_in[12];
    const float* bm  = (const float*)d_in[13];
    const float* al  = (const float*)d_in[14];
    const float* be  = (const float*)d_in[15];
    const float* g0  = (const float*)d_in[16];
    const float* b0  = (const float*)d_in[17];
    const float* Wl  = (const float*)d_in[18];
    const float* bl  = (const float*)d_in[19];
    const float* Wc  = (const float*)d_in[20];
    const float* bc  = (const float*)d_in[21];
    const float* g1  = (const float*)d_in[22];
    const float* b1  = (const float*)d_in[23];

    const size_t NL = (size_t)N_ * L_;          // 19200 tokens
    float* xp   = ws;                           // 4,915,200 f  (also reused for mm / conv out)
    float* cpb  = xp  + NL * HD_;               //   614,400 f
    float* disp = cpb + (size_t)N_ * S_ * 2 * HD_;  // 4,915,200 f (also reused for "new")
    float* y1   = disp + NL * HD_;              // 9,830,400 f
    // total workspace: 20,275,200 floats = ~81 MB

    const dim3 wave(32);

    // 1) xp = x @ Wp0 + bp0          (19200 x 256 x 256), 64x32 tiles
    gemm_wmma<4, 2, C_, C_, C_, HD_, HD_, C_, 0>
        <<<dim3(NL / 64, HD_ / 32), wave, 0, stream>>>(x, x, Wp0, bp0, xp);

    // 2) cp = center @ Wp1 + bp1     (1200 x 256 x 512)
    gemm_wmma_small<C_, 2 * HD_, 2 * HD_, C_>
        <<<dim3((N_ * S_) / 16, (2 * HD_) / 16), wave, 0, stream>>>(
            center, Wp1, bp1, cpb);

    // 3) similarity / argmax / dispatch
    sim_dispatch<<<dim3((L_ + 255) / 256, N_ * FC_), dim3(256), 0, stream>>>(
        xp, cpb, al, be, disp);

    // 4) mm = disp @ Wm + bm  -> reuse xp buffer
    gemm_wmma<4, 2, HD_, HD_, HD_, C_, C_, HD_, 0>
        <<<dim3(NL / 64, C_ / 32), wave, 0, stream>>>(disp, disp, Wm, bm, xp);

    // 5) new = LN(mm) * g0 + b0  -> reuse disp buffer
    layernorm_k<<<dim3((unsigned)NL), dim3(C_), 0, stream>>>(xp, g0, b0, nullptr, disp);

    // 6) y1 = gelu([x, new] @ Wl + bl)   (19200 x 512 x 512), virtual concat
    gemm_wmma<4, 2, C_, C_, C_, 2 * C_, 2 * C_, 2 * C_, 1>
        <<<dim3(NL / 64, (2 * C_) / 32), wave, 0, stream>>>(x, disp, Wl, bl, y1);

    // 7) conv3x3 SAME (512 -> 256) -> reuse xp buffer
    conv3x3_wmma_f32<<<dim3(NL / 16, C_ / 64), wave, 0, stream>>>(
        y1, Wc, bc, xp, hp, wp);

    // 8) out = LN(conv) * g1 + b1 + x
    layernorm_k<<<dim3((unsigned)NL), dim3(C_), 0, stream>>>(xp, g1, b1, x, out_slice);
}

extern "C" void kernel_launch(void* const* d_in, const int* in_sizes, int n_in,
                              void* d_out, int out_size, void* d_ws, size_t ws_size,
                              hipStream_t stream) {
    (void)in_sizes; (void)n_in; (void)out_size; (void)ws_size;

    const float* x0 = (const float*)d_in[0];
    const float* x1 = (const float*)d_in[1];
    const float* c0 = (const float*)d_in[2];
    const float* c1 = (const float*)d_in[3];
    const int*   h0 = (const int*)d_in[4];
    const int*   w0 = (const int*)d_in[5];
    const int*   h1 = (const int*)d_in[6];
    const int*   w1 = (const int*)d_in[7];

    float* out = (float*)d_out;
    float* ws  = (float*)d_ws;
    const size_t slice = (size_t)N_ * L_ * C_;

    // block 0: x0 with center1 ; block 1: x1 with center0 (shared params)
    run_block(x0, c1, h0, w0, d_in, out,         ws, stream);
    run_block(x1, c0, h1, w1, d_in, out + slice, ws, stream);
}